// VQVAE_19765439496688
// MI455X (gfx1250) — compile-verified
//
#include <hip/hip_runtime.h>
#include <hip/hip_bf16.h>
#include <math.h>

typedef float v2f __attribute__((ext_vector_type(2)));
typedef float v8f __attribute__((ext_vector_type(8)));

#if defined(__has_builtin)
#if __has_builtin(__builtin_amdgcn_wmma_f32_16x16x4_f32)
#define HAVE_WMMA_F32X4 1
#endif
#endif

// ---------------- flattened parameter indices (jax tree order: sorted dict keys) ----
// 0: inp
// dec: conv_in(w1,b2) conv_out(w3,b4) mid(b1 5,b2 6,bw1 7,bw2 8,g1 9,g2 10,w1 11,w2 12)
//      norm_out(g13,b14) res0(b1 15,b2 16,bw1 17,bw2 18,g1 19,g2 20,w1 21,w2 22)
//      res1(b1 23,b2 24,bw1 25,bw2 26,g1 27,g2 28,w1 29,w2 30) up0(w31,b32) up1(w33,b34)
// 35: embedding [512,8]
// enc: conv_in(w36,b37) conv_out(w38,b39) down0(w40,b41) down1(w42,b43)
//      mid(b1 44,b2 45,bw1 46,bw2 47,g1 48,g2 49,w1 50,w2 51) norm_out(g52,b53)
//      res0(b1 54,b2 55,bs 56,bw1 57,bw2 58,g1 59,g2 60,w1 61,w2 62,ws 63)
//      res1(b1 64,b2 65,bs 66,bw1 67,bw2 68,g1 69,g2 70,w1 71,w2 72,ws 73)
// phis: (w74,b75)(w76,b77)(w78,b79)(w80,b81)
// 82,83: post_quant_conv w,b    84,85: quant_conv w,b
struct P86 { const float* p[86]; };

struct ResPtrs {
  const float *g1,*b1,*w1,*bw1,*g2,*b2,*w2,*bw2,*ws,*bs;
};

// 3x3 conv with W=1 => only middle kernel column matters. Stride S along H.
template<int CI,int CO,int HI,int S>
__device__ __forceinline__ void conv3(const float (&in)[CI][HI], float (&out)[CO][HI/S],
                                      const float* __restrict__ w, const float* __restrict__ b) {
  constexpr int HO = HI / S;
  #pragma unroll
  for (int co = 0; co < CO; ++co) {
    float acc[HO];
    #pragma unroll
    for (int i = 0; i < HO; ++i) acc[i] = b[co];
    #pragma unroll
    for (int ci = 0; ci < CI; ++ci) {
      #pragma unroll
      for (int kh = 0; kh < 3; ++kh) {
        const float wv = w[((co*CI + ci)*3 + kh)*3 + 1];
        #pragma unroll
        for (int i = 0; i < HO; ++i) {
          const int src = i*S + kh - 1;
          if (src >= 0 && src < HI) acc[i] += wv * in[ci][src];
        }
      }
    }
    #pragma unroll
    for (int i = 0; i < HO; ++i) out[co][i] = acc[i];
  }
}

// GroupNorm(groups=2) + SiLU, in place
template<int C,int H>
__device__ __forceinline__ void gn_silu_ip(float (&x)[C][H], const float* __restrict__ g,
                                           const float* __restrict__ b) {
  constexpr int GS = C/2;
  #pragma unroll
  for (int grp = 0; grp < 2; ++grp) {
    float s = 0.f, s2 = 0.f;
    #pragma unroll
    for (int c = grp*GS; c < (grp+1)*GS; ++c) {
      #pragma unroll
      for (int i = 0; i < H; ++i) { float v = x[c][i]; s += v; s2 += v*v; }
    }
    const float invn = 1.0f / (float)(GS*H);
    const float m = s * invn;
    const float var = s2 * invn - m*m;
    const float r = rsqrtf(var + 1e-6f);
    #pragma unroll
    for (int c = grp*GS; c < (grp+1)*GS; ++c) {
      const float sc = g[c]*r, sh = b[c];
      #pragma unroll
      for (int i = 0; i < H; ++i) {
        float y = (x[c][i]-m)*sc + sh;
        x[c][i] = y / (1.0f + __expf(-y));
      }
    }
  }
}

template<int CI,int CO,int H,bool SKIP>
__device__ __forceinline__ void resblock(const float (&in)[CI][H], float (&out)[CO][H],
                                         const ResPtrs rp) {
  static_assert(SKIP || CI == CO, "identity skip needs CI==CO");
  float t[CI][H];
  #pragma unroll
  for (int c = 0; c < CI; ++c)
    #pragma unroll
    for (int i = 0; i < H; ++i) t[c][i] = in[c][i];
  gn_silu_ip<CI,H>(t, rp.g1, rp.b1);
  float a[CO][H];
  conv3<CI,CO,H,1>(t, a, rp.w1, rp.bw1);
  gn_silu_ip<CO,H>(a, rp.g2, rp.b2);
  conv3<CO,CO,H,1>(a, out, rp.w2, rp.bw2);
  if constexpr (SKIP) {
    #pragma unroll
    for (int co = 0; co < CO; ++co) {
      #pragma unroll
      for (int i = 0; i < H; ++i) {
        float s = rp.bs[co];
        #pragma unroll
        for (int ci = 0; ci < CI; ++ci) s += rp.ws[co*CI+ci]*in[ci][i];
        out[co][i] += s;
      }
    }
  } else {
    #pragma unroll
    for (int c = 0; c < CO; ++c)
      #pragma unroll
      for (int i = 0; i < H; ++i) out[c][i] += in[c][i];
  }
}

// ------------------------------ prep: normalized codebook, zero loss ----------------
__global__ void __launch_bounds__(256) k_prep(P86 prm, float* __restrict__ En,
                                              float* __restrict__ loss_slot) {
  const int gid = blockIdx.x*blockDim.x + threadIdx.x;
  if (gid == 0) *loss_slot = 0.0f;
  if (gid < 512) {
    const float* E = prm.p[35] + gid*8;
    float s = 0.f;
    #pragma unroll
    for (int c = 0; c < 8; ++c) s += E[c]*E[c];
    const float inv = 1.0f / fmaxf(sqrtf(s), 1e-12f);
    #pragma unroll
    for (int c = 0; c < 8; ++c) En[gid*8+c] = E[c]*inv;
  }
}

// ------------------------------ encoder: one thread per item ------------------------
__global__ void __launch_bounds__(256) k_encode(P86 prm, float* __restrict__ fbuf) {
  const size_t item = (size_t)blockIdx.x*blockDim.x + threadIdx.x;
  const float* inp = prm.p[0] + item*16;
  float x[1][16];
  #pragma unroll
  for (int i = 0; i < 16; ++i) x[0][i] = inp[i];

  float e0[2][16]; conv3<1,2,16,1>(x, e0, prm.p[36], prm.p[37]);
  const ResPtrs er0{prm.p[59],prm.p[54],prm.p[61],prm.p[57],prm.p[60],prm.p[55],prm.p[62],prm.p[58],prm.p[63],prm.p[56]};
  float r0[4][16]; resblock<2,4,16,true>(e0, r0, er0);
  float d0[4][8];  conv3<4,4,16,2>(r0, d0, prm.p[40], prm.p[41]);
  const ResPtrs er1{prm.p[69],prm.p[64],prm.p[71],prm.p[67],prm.p[70],prm.p[65],prm.p[72],prm.p[68],prm.p[73],prm.p[66]};
  float r1[8][8];  resblock<4,8,8,true>(d0, r1, er1);
  float d1[8][4];  conv3<8,8,8,2>(r1, d1, prm.p[42], prm.p[43]);
  const ResPtrs emid{prm.p[48],prm.p[44],prm.p[50],prm.p[46],prm.p[49],prm.p[45],prm.p[51],prm.p[47],nullptr,nullptr};
  float m[8][4];   resblock<8,8,4,false>(d1, m, emid);
  gn_silu_ip<8,4>(m, prm.p[52], prm.p[53]);
  float co[8][4];  conv3<8,8,4,1>(m, co, prm.p[38], prm.p[39]);
  float f[8][4];   conv3<8,8,4,1>(co, f, prm.p[84], prm.p[85]);   // quant_conv fused

  float* o = fbuf + item*32;
  #pragma unroll
  for (int c = 0; c < 8; ++c)
    #pragma unroll
    for (int h = 0; h < 4; ++h) o[c*4+h] = f[c][h];
}

// ------------------------------ one VQ scale (wave-cooperative WMMA argmax) ---------
// A 16x4 f32 layout: lanes 0-15 rows M=0..15 (V0=K0,V1=K1); lanes 16-31 same rows (V0=K2,V1=K3).
// B 4x16 f32 layout (mirror): lanes hold N=0..15; V0=K0|K2, V1=K1|K3 by lane half.
// C/D: VGPR r, lanes 0-15 -> M=r, N=lane; lanes 16-31 -> M=8+r, N=lane-16.
template<int PN>
__device__ __forceinline__ void quant_scale(
    float (&fr)[8][4], float& sq,
    float* __restrict__ sZw, int* __restrict__ sIdxw, const float* __restrict__ sEn,
    const float* __restrict__ E, const float* __restrict__ phw, const float* __restrict__ phb,
    const int lane, const int half, const int nl)
{
  // 1) stage pooled residual vectors (area pool H=4 -> PN); znorm skipped (argmax invariant)
  #pragma unroll
  for (int p = 0; p < PN; ++p) {
    #pragma unroll
    for (int c = 0; c < 8; ++c) {
      float v;
      if constexpr (PN == 1)      v = 0.25f*(fr[c][0]+fr[c][1]+fr[c][2]+fr[c][3]);
      else if constexpr (PN == 2) v = 0.5f*(fr[c][2*p]+fr[c][2*p+1]);
      else if constexpr (PN == 3) v = 0.5f*(fr[c][p]+fr[c][p+1]);
      else                        v = fr[c][p];
      sZw[(lane*PN + p)*8 + c] = v;
    }
  }
  __syncthreads();

  // 2) sims = Z @ En^T via WMMA f32 16x16x4 (2 x K=4), argmax over 512 codes
  #pragma unroll
  for (int mt = 0; mt < 2*PN; ++mt) {
    const float* zb = &sZw[(mt*16 + nl)*8 + 2*half];
    const v2f a0 = { zb[0], zb[1] };
    const v2f a1 = { zb[4], zb[5] };
    float bestv[8]; int besti[8];
    #pragma unroll
    for (int r = 0; r < 8; ++r) { bestv[r] = -3.4e38f; besti[r] = 0; }
    #pragma unroll 4
    for (int nt = 0; nt < 32; ++nt) {
      const float* eb = &sEn[(nt*16 + nl)*8 + 2*half];
      v8f cc = {0.f,0.f,0.f,0.f,0.f,0.f,0.f,0.f};
#ifdef HAVE_WMMA_F32X4
      const v2f b0 = { eb[0], eb[1] };
      const v2f b1 = { eb[4], eb[5] };
      cc = __builtin_amdgcn_wmma_f32_16x16x4_f32(false, a0, false, b0, (short)0, cc, false, false);
      cc = __builtin_amdgcn_wmma_f32_16x16x4_f32(false, a1, false, b1, (short)0, cc, false, false);
#else
      // VALU fallback emulating the same C layout
      #pragma unroll
      for (int r = 0; r < 8; ++r) {
        const float* zr = &sZw[(mt*16 + 8*half + r)*8];
        const float* er = &sEn[(nt*16 + nl)*8];
        float s = 0.f;
        #pragma unroll
        for (int k = 0; k < 8; ++k) s += zr[k]*er[k];
        cc[r] = s;
      }
#endif
      const int code = nt*16 + nl;
      #pragma unroll
      for (int r = 0; r < 8; ++r) {
        const float v = cc[r];
        if (v > bestv[r]) { bestv[r] = v; besti[r] = code; }   // branchless cndmask pair
      }
    }
    // cross-lane argmax within each 16-lane half, branchless:
    // pack (monotonic(val) << 32) | (511 - idx); u64 max == (max val, then min idx),
    // matching jnp.argmax first-max tie-break.
    #pragma unroll
    for (int r = 0; r < 8; ++r) {
      unsigned int ub = __float_as_uint(bestv[r]);
      ub ^= (unsigned int)((int)ub >> 31) | 0x80000000u;
      unsigned long long key = ((unsigned long long)ub << 32) | (unsigned int)(511 - besti[r]);
      #pragma unroll
      for (int s = 1; s <= 8; s <<= 1) {
        const unsigned int klo = (unsigned int)key;
        const unsigned int khi = (unsigned int)(key >> 32);
        const unsigned int olo = (unsigned int)__shfl_xor((int)klo, s, 32);
        const unsigned int ohi = (unsigned int)__shfl_xor((int)khi, s, 32);
        const unsigned long long ok = ((unsigned long long)ohi << 32) | olo;
        key = (ok > key) ? ok : key;
      }
      if (nl == 0) sIdxw[mt*16 + half*8 + r] = 511 - (int)((unsigned int)key & 0x1FFu);
    }
  }
  __syncthreads();

  // 3) gather codes, bicubic upsample, Phi partial-residual conv, residual update
  float hu[8][4];
  {
    float hs[PN][8];
    #pragma unroll
    for (int p = 0; p < PN; ++p) {
      const int id = sIdxw[lane*PN + p];
      #pragma unroll
      for (int c = 0; c < 8; ++c) hs[p][c] = E[id*8 + c];
    }
    if constexpr (PN == 1) {
      #pragma unroll
      for (int c = 0; c < 8; ++c)
        #pragma unroll
        for (int h = 0; h < 4; ++h) hu[c][h] = hs[0][c];
    } else if constexpr (PN == 2) {
      const float U[4][2] = {{ 1.08823529f,-0.08823529f},{0.79285714f,0.20714286f},
                             { 0.20714286f, 0.79285714f},{-0.08823529f,1.08823529f}};
      #pragma unroll
      for (int c = 0; c < 8; ++c)
        #pragma unroll
        for (int h = 0; h < 4; ++h) hu[c][h] = U[h][0]*hs[0][c] + U[h][1]*hs[1][c];
    } else if constexpr (PN == 3) {
      const float U[4][3] = {{ 1.05223881f,-0.05223881f, 0.f},
                             { 0.37324602f, 0.69691300f,-0.07015903f},
                             {-0.07015903f, 0.69691300f, 0.37324602f},
                             { 0.f,-0.05223881f, 1.05223881f}};
      #pragma unroll
      for (int c = 0; c < 8; ++c)
        #pragma unroll
        for (int h = 0; h < 4; ++h)
          hu[c][h] = U[h][0]*hs[0][c] + U[h][1]*hs[1][c] + U[h][2]*hs[2][c];
    } else {
      #pragma unroll
      for (int c = 0; c < 8; ++c)
        #pragma unroll
        for (int h = 0; h < 4; ++h) hu[c][h] = hs[h][c];
    }
  }
  float hm[8][4];
  conv3<8,8,4,1>(hu, hm, phw, phb);
  #pragma unroll
  for (int c = 0; c < 8; ++c) {
    #pragma unroll
    for (int h = 0; h < 4; ++h) {
      fr[c][h] -= 0.5f*hu[c][h] + 0.5f*hm[c][h];
      sq += fr[c][h]*fr[c][h];          // == (f_hat - f)^2 for this scale
    }
  }
}

// ------------------------------ quantizer kernel ------------------------------------
__global__ void __launch_bounds__(256) k_quant(P86 prm, const float* __restrict__ EnG,
                                               float* __restrict__ fbuf,
                                               float* __restrict__ loss_slot,
                                               const float lscale) {
  __shared__ float sEn[512*8];     // normalized codebook (16 KB)
  __shared__ float sZ[8][1024];    // per-wave staged residual vectors (32 KB)
  __shared__ int   sIdx[8][128];   // per-wave argmax indices (4 KB)
  __shared__ float sRed[256];

  const int tid = threadIdx.x;
  for (int i = tid; i < 512*8; i += 256) sEn[i] = EnG[i];
  __syncthreads();

  const size_t item = (size_t)blockIdx.x*256 + tid;
  const int wave = tid >> 5, lane = tid & 31, half = lane >> 4, nl = lane & 15;

  float f[8][4], fr[8][4];
  {
    const float* src = fbuf + item*32;
    #pragma unroll
    for (int c = 0; c < 8; ++c)
      #pragma unroll
      for (int h = 0; h < 4; ++h) { f[c][h] = src[c*4+h]; fr[c][h] = f[c][h]; }
  }

  float sq = 0.f;
  const float* E = prm.p[35];
  quant_scale<1>(fr, sq, sZ[wave], sIdx[wave], sEn, E, prm.p[74], prm.p[75], lane, half, nl);
  quant_scale<2>(fr, sq, sZ[wave], sIdx[wave], sEn, E, prm.p[76], prm.p[77], lane, half, nl);
  quant_scale<3>(fr, sq, sZ[wave], sIdx[wave], sEn, E, prm.p[78], prm.p[79], lane, half, nl);
  quant_scale<4>(fr, sq, sZ[wave], sIdx[wave], sEn, E, prm.p[80], prm.p[81], lane, half, nl);

  {
    float* dst = fbuf + item*32;                    // f_hat = f - f_rest (in place)
    #pragma unroll
    for (int c = 0; c < 8; ++c)
      #pragma unroll
      for (int h = 0; h < 4; ++h) dst[c*4+h] = f[c][h] - fr[c][h];
  }

  sRed[tid] = sq;
  __syncthreads();
  #pragma unroll
  for (int s = 128; s > 0; s >>= 1) {
    if (tid < s) sRed[tid] += sRed[tid + s];
    __syncthreads();
  }
  if (tid == 0) atomicAdd(loss_slot, sRed[0] * lscale);
}

// ------------------------------ decoder: one thread per item ------------------------
__global__ void __launch_bounds__(256) k_decode(P86 prm, const float* __restrict__ fbuf,
                                                float* __restrict__ out) {
  const size_t item = (size_t)blockIdx.x*blockDim.x + threadIdx.x;
  float fh[8][4];
  {
    const float* src = fbuf + item*32;
    #pragma unroll
    for (int c = 0; c < 8; ++c)
      #pragma unroll
      for (int h = 0; h < 4; ++h) fh[c][h] = src[c*4+h];
  }
  float g[8][4];  conv3<8,8,4,1>(fh, g, prm.p[82], prm.p[83]);    // post_quant_conv
  float h0[8][4]; conv3<8,8,4,1>(g, h0, prm.p[1], prm.p[2]);      // dec conv_in
  const ResPtrs dmid{prm.p[9],prm.p[5],prm.p[11],prm.p[7],prm.p[10],prm.p[6],prm.p[12],prm.p[8],nullptr,nullptr};
  float hm[8][4]; resblock<8,8,4,false>(h0, hm, dmid);

  float u1i[8][8];
  #pragma unroll
  for (int c = 0; c < 8; ++c)
    #pragma unroll
    for (int i = 0; i < 8; ++i) u1i[c][i] = hm[c][i>>1];          // repeat x2
  float u1[4][8]; conv3<8,4,8,1>(u1i, u1, prm.p[33], prm.p[34]);  // up1
  const ResPtrs dr1{prm.p[27],prm.p[23],prm.p[29],prm.p[25],prm.p[28],prm.p[24],prm.p[30],prm.p[26],nullptr,nullptr};
  float rr1[4][8]; resblock<4,4,8,false>(u1, rr1, dr1);

  float u0i[4][16];
  #pragma unroll
  for (int c = 0; c < 4; ++c)
    #pragma unroll
    for (int i = 0; i < 16; ++i) u0i[c][i] = rr1[c][i>>1];        // repeat x2
  float u0[2][16]; conv3<4,2,16,1>(u0i, u0, prm.p[31], prm.p[32]); // up0
  const ResPtrs dr0{prm.p[19],prm.p[15],prm.p[21],prm.p[17],prm.p[20],prm.p[16],prm.p[22],prm.p[18],nullptr,nullptr};
  float rr0[2][16]; resblock<2,2,16,false>(u0, rr0, dr0);

  gn_silu_ip<2,16>(rr0, prm.p[13], prm.p[14]);
  float y[1][16]; conv3<2,1,16,1>(rr0, y, prm.p[3], prm.p[4]);

  float* o = out + item*16;
  #pragma unroll
  for (int i = 0; i < 16; ++i) o[i] = y[0][i];
}

// ------------------------------ host launcher ---------------------------------------
extern "C" void kernel_launch(void* const* d_in, const int* in_sizes, int n_in,
                              void* d_out, int out_size, void* d_ws, size_t ws_size,
                              hipStream_t stream) {
  P86 prm;
  for (int i = 0; i < 86; ++i) prm.p[i] = (const float*)d_in[i < n_in ? i : (n_in-1)];

  const int B = in_sizes[0] / 16;                    // 65536
  float* En   = (float*)d_ws;                        // 512*8 normalized codebook
  float* fbuf = En + 512*8;                          // B*32: f, then f_hat in place
  float* out  = (float*)d_out;
  float* loss_slot = out + (out_size - 1);

  const float lscale = (float)(1.25 / (4.0 * (double)B * 32.0)); // (1+beta)/SN / (B*C*H*W)

  k_prep  <<<2,      256, 0, stream>>>(prm, En, loss_slot);
  k_encode<<<B/256,  256, 0, stream>>>(prm, fbuf);
  k_quant <<<B/256,  256, 0, stream>>>(prm, En, fbuf, loss_slot, lscale);
  k_decode<<<B/256,  256, 0, stream>>>(prm, fbuf, out);
}